// MyGraphConvModel_42116449305180
// MI455X (gfx1250) — compile-verified
//
#include <hip/hip_runtime.h>
#include <math.h>

// ---------------------------------------------------------------------------
// MI455X / gfx1250 implementation of the DeepChem GraphConvModel reference.
// GEMMs run on V_WMMA_F32_16X16X4_F32 (exact fp32). Gathers stay fp32 and
// L2-resident. dense1 + tanh + BN + segment-sum/max are fused (atomics) so no
// [N,256] intermediate is ever materialized.
// ---------------------------------------------------------------------------

typedef __attribute__((ext_vector_type(2))) float v2f;
typedef __attribute__((ext_vector_type(8))) float v8f;

#define N_ATOMS 131072
#define BATCH   4096
#define NTASKS  12

__constant__ int kStarts[12] = {0, 2048, 34816, 75776, 108544, 129024,
                                130048, 130560, 130816, 130944, 131008, 131072};

struct AdjPtrs { const int* p[10]; };

__device__ __forceinline__ float seluf(float x) {
    const float scale = 1.0507009873554805f;
    const float alpha = 1.6732632423543772f;
    return scale * (x > 0.f ? x : alpha * (expf(x) - 1.f));
}

__device__ __forceinline__ v8f vzero8() {
    v8f z;
    #pragma unroll
    for (int i = 0; i < 8; ++i) z[i] = 0.f;
    return z;
}

// Concatenated weight read: rows [0,FI) -> W_self[d], rows [FI,2FI) -> W_rel[d-1].
template<int FI, int FO>
__device__ __forceinline__ float wmat(const float* __restrict__ Ws,
                                      const float* __restrict__ Wr,
                                      int k, int n) {
    if (FO % 16 != 0 && n >= FO) return 0.f;
    if (k < FI) return Ws[k * FO + n];
    int k2 = k - FI;
    if (k2 < FI) return Wr[k2 * FO + n];
    return 0.f;   // K padding
}

// float atomic-max via monotone int/uint trick (works with -inf init, mixed signs)
__device__ __forceinline__ void atomicMaxF(float* addr, float val) {
    if (val >= 0.f) atomicMax((int*)addr, __float_as_int(val));
    else            atomicMin((unsigned int*)addr, __float_as_uint(val));
}

// ---------------------------------------------------------------------------
// Graph convolution layer: one wave per 16-atom tile (tiles never cross a
// degree boundary: all DEG_COUNTS are multiples of 64).
//   out[16,FO] = selu( [x_self | nbr_sum] (16 x 2FI)  @  [[Ws[d]];[Wr[d-1]]] + bias )
// ---------------------------------------------------------------------------
template<int FI, int FO>
__global__ void __launch_bounds__(128) conv_layer(
    const float* __restrict__ xin,   // [N, FI]
    float*       __restrict__ xout,  // [N, FO]
    const float* __restrict__ ws,    // [11, FI, FO]
    const float* __restrict__ wr,    // [10, FI, FO]
    const float* __restrict__ bs,    // [11, FO]
    const float* __restrict__ br,    // [10, FO]
    AdjPtrs adj)
{
    constexpr int KC  = 2 * FI;              // concat K
    constexpr int KCP = (KC + 3) & ~3;       // K padded to multiple of 4
    constexpr int STR = KCP + 4;             // LDS row stride (bank-conflict dodge)
    constexpr int NT  = (FO + 15) / 16;      // N tiles of 16

    __shared__ float sm[4][16 * STR];

    const int wave = threadIdx.x >> 5;
    const int lane = threadIdx.x & 31;
    const int tile = blockIdx.x * 4 + wave;
    const int row0 = tile * 16;

    // degree of this tile (wave-uniform)
    int d = 0;
    #pragma unroll
    for (int i = 1; i <= 10; ++i) d += (row0 >= kStarts[i]) ? 1 : 0;

    float* smw = &sm[wave][0];
    const int r = lane & 15;

    // ---- stage [x_self | neighbor-sum] into LDS (2 lanes per row) ----
    {
        const int half = lane >> 4;
        const int f0 = half ? (FI + 1) / 2 : 0;
        const int f1 = half ? FI : (FI + 1) / 2;
        const int atom = row0 + r;
        const float* xa = xin + (size_t)atom * FI;
        float* dst = smw + r * STR;
        for (int f = f0; f < f1; ++f) dst[f] = xa[f];
        if (d > 0) {
            const int* a = adj.p[d - 1] + (size_t)(atom - kStarts[d]) * d;
            int nb[10];
            for (int j = 0; j < d; ++j) nb[j] = a[j];
            for (int f = f0; f < f1; ++f) {
                float s = 0.f;
                for (int j = 0; j < d; ++j) s += xin[(size_t)nb[j] * FI + f];
                dst[FI + f] = s;
            }
        } else {
            for (int f = f0; f < f1; ++f) dst[FI + f] = 0.f;
        }
        if (half) {   // zero K padding (read by WMMA A fragments)
            #pragma unroll
            for (int k = KC; k < KCP; ++k) dst[k] = 0.f;
        }
    }
    __builtin_amdgcn_wave_barrier();   // per-wave LDS: DS ops are in-order

    // ---- fused self+rel GEMM on V_WMMA_F32_16X16X4_F32 ----
    const float* Ws = ws + (size_t)d * (FI * FO);
    const float* Wr = wr + (size_t)((d > 0 ? d : 1) - 1) * (FI * FO);

    v8f acc[NT];
    #pragma unroll
    for (int t = 0; t < NT; ++t) acc[t] = vzero8();

    const int kro = (lane < 16) ? 0 : 2;     // A/B lane K-offset (ISA 7.12.2)
    const float* arow = smw + r * STR;

    for (int kb = 0; kb < KCP; kb += 4) {
        v2f a;
        a.x = arow[kb + kro];
        a.y = arow[kb + kro + 1];
        #pragma unroll
        for (int t = 0; t < NT; ++t) {
            const int n  = t * 16 + r;
            const int k0 = kb + kro;
            v2f b;
            b.x = wmat<FI, FO>(Ws, Wr, k0,     n);
            b.y = wmat<FI, FO>(Ws, Wr, k0 + 1, n);
            acc[t] = __builtin_amdgcn_wmma_f32_16x16x4_f32(
                false, a, false, b, (short)0, acc[t], false, false);
        }
    }

    // ---- bias + SELU epilogue (C/D layout: VGPR v -> rows v / v+8) ----
    const int mbase = (lane < 16) ? 0 : 8;
    #pragma unroll
    for (int t = 0; t < NT; ++t) {
        const int n = t * 16 + r;
        if (n < FO) {
            float bias = bs[d * FO + n] + ((d > 0) ? br[(d - 1) * FO + n] : 0.f);
            #pragma unroll
            for (int v = 0; v < 8; ++v) {
                float val = acc[t][v] + bias;
                xout[(size_t)(row0 + mbase + v) * FO + n] = seluf(val);
            }
        }
    }
}

// ---------------------------------------------------------------------------
// graph_pool: out[i] = max(x[i], max over neighbors), F = 36
// ---------------------------------------------------------------------------
__global__ void __launch_bounds__(256) pool_max(
    const float* __restrict__ xin, float* __restrict__ xout, AdjPtrs adj)
{
    const int atom = blockIdx.x * 256 + threadIdx.x;
    if (atom >= N_ATOMS) return;
    int d = 0;
    #pragma unroll
    for (int i = 1; i <= 10; ++i) d += (atom >= kStarts[i]) ? 1 : 0;

    float v[36];
    const float* xa = xin + (size_t)atom * 36;
    #pragma unroll
    for (int f = 0; f < 36; ++f) v[f] = xa[f];
    if (d > 0) {
        const int* a = adj.p[d - 1] + (size_t)(atom - kStarts[d]) * d;
        for (int j = 0; j < d; ++j) {
            const float* xn = xin + (size_t)a[j] * 36;
            #pragma unroll
            for (int f = 0; f < 36; ++f) v[f] = fmaxf(v[f], xn[f]);
        }
    }
    float* o = xout + (size_t)atom * 36;
    #pragma unroll
    for (int f = 0; f < 36; ++f) o[f] = v[f];
}

// ---------------------------------------------------------------------------
// init segment accumulators: S = 0, M = -inf
// ---------------------------------------------------------------------------
__global__ void init_segments(float* __restrict__ S, float* __restrict__ Mx) {
    const int i = blockIdx.x * 256 + threadIdx.x;
    if (i < BATCH * 256) { S[i] = 0.f; Mx[i] = -__builtin_inff(); }
}

// ---------------------------------------------------------------------------
// dense1 (36->256, WMMA) + tanh + inference BN, fused with segment sum/max
// scatter (atomics). Block = 16 rows x 256 cols; wave w owns 32 cols.
// ---------------------------------------------------------------------------
__global__ void __launch_bounds__(256) dense1_bn_gather(
    const float* __restrict__ x,     // pooled [N, 36]
    const float* __restrict__ W,     // [36, 256]
    const float* __restrict__ bvec,  // [256]
    const float* __restrict__ gam, const float* __restrict__ bet,
    const float* __restrict__ mu,  const float* __restrict__ var,
    const int*   __restrict__ membership,
    float* __restrict__ S, float* __restrict__ Mx)
{
    __shared__ float A[16 * 40];
    const int row0 = blockIdx.x * 16;

    for (int i = threadIdx.x; i < 16 * 36; i += 256) {
        const int rr = i / 36, ff = i % 36;
        A[rr * 40 + ff] = x[(size_t)(row0 + rr) * 36 + ff];
    }
    __syncthreads();

    const int wave = threadIdx.x >> 5;
    const int lane = threadIdx.x & 31;
    const int r    = lane & 15;
    const int kro  = (lane < 16) ? 0 : 2;
    const int c0   = wave * 32 + r;
    const int c1   = c0 + 16;

    v8f acc0 = vzero8(), acc1 = vzero8();
    #pragma unroll
    for (int kb = 0; kb < 36; kb += 4) {     // K = 36, exact multiple of 4
        v2f a;
        a.x = A[r * 40 + kb + kro];
        a.y = A[r * 40 + kb + kro + 1];
        v2f b0, b1;
        b0.x = W[(kb + kro) * 256 + c0];  b0.y = W[(kb + kro + 1) * 256 + c0];
        b1.x = W[(kb + kro) * 256 + c1];  b1.y = W[(kb + kro + 1) * 256 + c1];
        acc0 = __builtin_amdgcn_wmma_f32_16x16x4_f32(false, a, false, b0, (short)0, acc0, false, false);
        acc1 = __builtin_amdgcn_wmma_f32_16x16x4_f32(false, a, false, b1, (short)0, acc1, false, false);
    }

    const int mb = (lane < 16) ? 0 : 8;
    const float s0 = rsqrtf(var[c0] + 1e-3f) * gam[c0];
    const float s1 = rsqrtf(var[c1] + 1e-3f) * gam[c1];
    const float d0 = bvec[c0], d1 = bvec[c1];
    const float m0 = mu[c0], m1 = mu[c1];
    const float e0 = bet[c0], e1 = bet[c1];

    #pragma unroll
    for (int v = 0; v < 8; ++v) {
        const int atom = row0 + mb + v;
        const int mol  = membership[atom];
        const float h0 = (tanhf(acc0[v] + d0) - m0) * s0 + e0;
        const float h1 = (tanhf(acc1[v] + d1) - m1) * s1 + e1;
        atomicAdd(&S[(size_t)mol * 256 + c0], h0);
        atomicAdd(&S[(size_t)mol * 256 + c1], h1);
        atomicMaxF(&Mx[(size_t)mol * 256 + c0], h0);
        atomicMaxF(&Mx[(size_t)mol * 256 + c1], h1);
    }
}

// ---------------------------------------------------------------------------
// head: g = tanh([S|M]) [512]; logits = g @ W2 + b2 [24]; pairwise softmax.
// One wave per molecule.
// ---------------------------------------------------------------------------
__global__ void __launch_bounds__(32) head_kernel(
    const float* __restrict__ S, const float* __restrict__ Mx,
    const float* __restrict__ W2,   // [512, 24]
    const float* __restrict__ b2,   // [24]
    float* __restrict__ out)        // [BATCH, 12, 2]
{
    __shared__ float g[512];
    __shared__ float logit[24];
    const int m    = blockIdx.x;
    const int lane = threadIdx.x;

    for (int i = lane; i < 256; i += 32) {
        g[i]       = tanhf(S[(size_t)m * 256 + i]);
        g[256 + i] = tanhf(Mx[(size_t)m * 256 + i]);
    }
    __syncthreads();

    if (lane < 24) {
        float acc = b2[lane];
        for (int k = 0; k < 512; ++k) acc += g[k] * W2[k * 24 + lane];
        logit[lane] = acc;
    }
    __syncthreads();

    if (lane < 12) {
        const float l0 = logit[2 * lane], l1 = logit[2 * lane + 1];
        const float mxv = fmaxf(l0, l1);
        const float p0 = expf(l0 - mxv), p1 = expf(l1 - mxv);
        const float inv = 1.f / (p0 + p1);
        out[(size_t)m * 24 + 2 * lane]     = p0 * inv;
        out[(size_t)m * 24 + 2 * lane + 1] = p1 * inv;
    }
}

// ---------------------------------------------------------------------------
extern "C" void kernel_launch(void* const* d_in, const int* in_sizes, int n_in,
                              void* d_out, int out_size, void* d_ws, size_t ws_size,
                              hipStream_t stream) {
    (void)in_sizes; (void)n_in; (void)out_size; (void)ws_size;

    const float* atom_features = (const float*)d_in[0];
    const int*   membership    = (const int*)d_in[2];

    AdjPtrs adj;
    for (int d = 1; d <= 10; ++d) adj.p[d - 1] = (const int*)d_in[2 + d];

    const float* ws1 = (const float*)d_in[13];
    const float* wr1 = (const float*)d_in[14];
    const float* bs1 = (const float*)d_in[15];
    const float* br1 = (const float*)d_in[16];
    const float* ws2 = (const float*)d_in[17];
    const float* wr2 = (const float*)d_in[18];
    const float* bs2 = (const float*)d_in[19];
    const float* br2 = (const float*)d_in[20];
    const float* ws3 = (const float*)d_in[21];
    const float* wr3 = (const float*)d_in[22];
    const float* bs3 = (const float*)d_in[23];
    const float* br3 = (const float*)d_in[24];
    const float* ws4 = (const float*)d_in[25];
    const float* wr4 = (const float*)d_in[26];
    const float* bs4 = (const float*)d_in[27];
    const float* br4 = (const float*)d_in[28];
    const float* d1w = (const float*)d_in[29];
    const float* d1b = (const float*)d_in[30];
    const float* gam = (const float*)d_in[31];
    const float* bet = (const float*)d_in[32];
    const float* mu  = (const float*)d_in[33];
    const float* var = (const float*)d_in[34];
    const float* d2w = (const float*)d_in[35];
    const float* d2b = (const float*)d_in[36];

    // workspace: bufA[N*36] | bufB[N*36] | S[4096*256] | M[4096*256]  (~46 MB)
    float* bufA = (float*)d_ws;
    float* bufB = bufA + (size_t)N_ATOMS * 36;
    float* S    = bufB + (size_t)N_ATOMS * 36;
    float* Mx   = S    + (size_t)BATCH * 256;

    const dim3 convGrid(N_ATOMS / 16 / 4);   // 4 waves (tiles) per 128-thread block

    conv_layer<75, 15><<<convGrid, 128, 0, stream>>>(atom_features, bufA, ws1, wr1, bs1, br1, adj);
    conv_layer<15, 20><<<convGrid, 128, 0, stream>>>(bufA, bufB, ws2, wr2, bs2, br2, adj);
    conv_layer<20, 27><<<convGrid, 128, 0, stream>>>(bufB, bufA, ws3, wr3, bs3, br3, adj);
    conv_layer<27, 36><<<convGrid, 128, 0, stream>>>(bufA, bufB, ws4, wr4, bs4, br4, adj);

    pool_max<<<dim3(N_ATOMS / 256), 256, 0, stream>>>(bufB, bufA, adj);

    init_segments<<<dim3((BATCH * 256 + 255) / 256), 256, 0, stream>>>(S, Mx);

    dense1_bn_gather<<<dim3(N_ATOMS / 16), 256, 0, stream>>>(
        bufA, d1w, d1b, gam, bet, mu, var, membership, S, Mx);

    head_kernel<<<dim3(BATCH), 32, 0, stream>>>(S, Mx, d2w, d2b, (float*)d_out);
}